// SumGCNEncoder_37254546326128
// MI455X (gfx1250) — compile-verified
//
#include <hip/hip_runtime.h>
#include <hip/hip_bf16.h>

typedef __attribute__((ext_vector_type(2))) float v2f;
typedef __attribute__((ext_vector_type(8))) float v8f;

#define DIN  128
#define DOUT 128
#define NS   5
#define BM   32          // rows per block (two 16-row WMMA tiles per wave)

// ---------------------------------------------------------------------------
// Zero-fill (grid-stride, float4)
// ---------------------------------------------------------------------------
__global__ void zero_kernel(float* __restrict__ p, size_t n4) {
    size_t i = (size_t)blockIdx.x * blockDim.x + threadIdx.x;
    size_t stride = (size_t)gridDim.x * blockDim.x;
    float4* p4 = (float4*)p;
    for (; i < n4; i += stride) p4[i] = make_float4(0.f, 0.f, 0.f, 0.f);
}

// ---------------------------------------------------------------------------
// Cumulative weights, written directly in WMMA B-fragment order.
// weight layout [DIN][DOUT][NS]; output bsw[s] is [ntile(8)][kq(32)][lane(32)][2]:
//   lane = 16*h + lo holds B[4*kq + 2*h + j][16*nt + lo], j = 0,1
// so the GEMM reads one coalesced b64 per lane per K-step.
// ---------------------------------------------------------------------------
__global__ void cumw_kernel(const float* __restrict__ w, float* __restrict__ bsw) {
    int idx = blockIdx.x * blockDim.x + threadIdx.x;   // idx = k*DOUT + n
    if (idx >= DIN * DOUT) return;
    int k  = idx >> 7,  n  = idx & 127;
    int nt = n >> 4,    lo = n & 15;
    int kq = k >> 2,    kr = k & 3;
    int h  = kr >> 1,   j  = kr & 1;
    size_t off = ((((size_t)nt * 32 + kq) * 32 + (h * 16 + lo)) * 2 + j);
    float acc = 0.f;
#pragma unroll
    for (int s = 0; s < NS; ++s) {
        acc += w[(size_t)idx * NS + s];
        bsw[(size_t)s * DIN * DOUT + off] = acc;
    }
}

// ---------------------------------------------------------------------------
// WMMA fp32 GEMM: C[M,128] = A[M,128] @ B[128,128]  (B pre-swizzled)
// 256 threads = 8 waves; block tile 32 rows x 128 cols.
// A tile staged in LDS (coalesced float4 loads, +4 float row pad ->
// conflict-free b64 reads). Each wave: two accumulators (rows m0..15 and
// m0+16..31) sharing one B fragment -> 64 WMMAs / wave.
// ---------------------------------------------------------------------------
__global__ __launch_bounds__(256) void gemm_wmma_f32(const float* __restrict__ A,
                                                     const float* __restrict__ Bsw,
                                                     float* __restrict__ C,
                                                     int M) {
    __shared__ float As[BM][DIN + 4];          // 32 x 132 floats, 16.5 KB

    const int tid  = threadIdx.x;
    const int lane = tid & 31;
    const int wave = tid >> 5;
    const int m0   = blockIdx.x * BM;

    // stage A tile: 32x128 floats = 1024 float4, 4 per thread, coalesced
#pragma unroll
    for (int i = 0; i < 4; ++i) {
        int f4 = tid + i * 256;                // 0..1023
        int r  = f4 >> 5;                      // 32 float4 per row
        int c4 = f4 & 31;
        int gr = m0 + r; gr = gr < M ? gr : (M - 1);
        float4 v = *((const float4*)(A + (size_t)gr * DIN) + c4);
        *((float4*)&As[r][c4 * 4]) = v;
    }
    __syncthreads();

    const int lo = lane & 15;
    const int h  = lane >> 4;                  // K half select
    const float2* bp = (const float2*)Bsw + (size_t)wave * (32 * 32) + lane;

    v8f acc0 = {};
    v8f acc1 = {};
#pragma unroll
    for (int kq = 0; kq < 32; ++kq) {
        v2f a0 = *(const v2f*)&As[lo     ][kq * 4 + h * 2];
        v2f a1 = *(const v2f*)&As[lo + 16][kq * 4 + h * 2];
        float2 bf = bp[(size_t)kq * 32];
        v2f b; b.x = bf.x; b.y = bf.y;
        acc0 = __builtin_amdgcn_wmma_f32_16x16x4_f32(
            false, a0, false, b, (short)0, acc0, false, false);
        acc1 = __builtin_amdgcn_wmma_f32_16x16x4_f32(
            false, a1, false, b, (short)0, acc1, false, false);
    }

    const int n0 = wave * 16;
    {
        int crow = m0 + h * 8;
        float* cp = C + (size_t)crow * DOUT + n0 + lo;
#pragma unroll
        for (int i = 0; i < 8; ++i)
            if (crow + i < M) cp[(size_t)i * DOUT] = acc0[i];
    }
    {
        int crow = m0 + 16 + h * 8;
        float* cp = C + (size_t)crow * DOUT + n0 + lo;
#pragma unroll
        for (int i = 0; i < 8; ++i)
            if (crow + i < M) cp[(size_t)i * DOUT] = acc1[i];
    }
}

// ---------------------------------------------------------------------------
// COO SpMM scatter: out[rows[e], :] += vals[e] * dense[cols[e], :]
// one wave per edge; edge metadata scalarized via readfirstlane;
// float4 gather (L2-resident) + 4 fp32 HW atomics per lane.
// ---------------------------------------------------------------------------
__global__ __launch_bounds__(256) void spmm_kernel(const int*   __restrict__ rows,
                                                   const int*   __restrict__ cols,
                                                   const float* __restrict__ vals,
                                                   const float* __restrict__ dense,
                                                   float*       __restrict__ out,
                                                   int nE) {
    int gid  = blockIdx.x * blockDim.x + threadIdx.x;
    int lane = gid & 31;
    int e    = __builtin_amdgcn_readfirstlane(gid >> 5);   // wave-uniform -> SGPR
    if (e >= nE) return;

    int   r = rows[e];
    int   c = cols[e];
    float v = vals[e];

    const float4* src = (const float4*)(dense + (size_t)c * DOUT) + lane;
    float4 d = *src;

    float* dst = out + (size_t)r * DOUT + lane * 4;
    unsafeAtomicAdd(dst + 0, v * d.x);
    unsafeAtomicAdd(dst + 1, v * d.y);
    unsafeAtomicAdd(dst + 2, v * d.z);
    unsafeAtomicAdd(dst + 3, v * d.w);
}

// ---------------------------------------------------------------------------
// In-place ReLU
// ---------------------------------------------------------------------------
__global__ void relu_kernel(float* __restrict__ p, size_t n4) {
    size_t i = (size_t)blockIdx.x * blockDim.x + threadIdx.x;
    size_t stride = (size_t)gridDim.x * blockDim.x;
    float4* p4 = (float4*)p;
    for (; i < n4; i += stride) {
        float4 x = p4[i];
        x.x = x.x > 0.f ? x.x : 0.f;
        x.y = x.y > 0.f ? x.y : 0.f;
        x.z = x.z > 0.f ? x.z : 0.f;
        x.w = x.w > 0.f ? x.w : 0.f;
        p4[i] = x;
    }
}

// ---------------------------------------------------------------------------
// kernel_launch
// d_in: 0 user_inputs[NU,128] 1 item_inputs[NI,128] 2 weight[128,128,5]
//       3 u_rows[5,E] 4 u_cols[5,E] 5 u_vals[5,E]
//       6 i_rows[5,E] 7 i_cols[5,E] 8 i_vals[5,E]
// d_out: relu(user_hidden)[NU,128] ++ relu(item_hidden)[NI,128]
// ---------------------------------------------------------------------------
extern "C" void kernel_launch(void* const* d_in, const int* in_sizes, int n_in,
                              void* d_out, int out_size, void* d_ws, size_t ws_size,
                              hipStream_t stream) {
    const float* U      = (const float*)d_in[0];
    const float* V      = (const float*)d_in[1];
    const float* W      = (const float*)d_in[2];
    const int*   u_rows = (const int*)d_in[3];
    const int*   u_cols = (const int*)d_in[4];
    const float* u_vals = (const float*)d_in[5];
    const int*   i_rows = (const int*)d_in[6];
    const int*   i_cols = (const int*)d_in[7];
    const float* i_vals = (const float*)d_in[8];

    const int NU = in_sizes[0] / DIN;
    const int NI = in_sizes[1] / DIN;
    const int E  = in_sizes[5] / NS;

    float* out         = (float*)d_out;
    float* user_hidden = out;                        // [NU,128]
    float* item_hidden = out + (size_t)NU * DOUT;    // [NI,128]

    // workspace layout
    float* bsw   = (float*)d_ws;                             // [NS][8][32][32][2]
    float* tmp_u = bsw + (size_t)NS * DIN * DOUT;            // [NU,128]
    float* tmp_v = tmp_u + (size_t)NU * DOUT;                // [NI,128]

    // 1) zero hidden accumulators
    {
        size_t n4 = ((size_t)(NU + NI) * DOUT) / 4;
        int blocks = (int)((n4 + 255) / 256);
        if (blocks > 4096) blocks = 4096;
        zero_kernel<<<blocks, 256, 0, stream>>>(out, n4);
    }

    // 2) cumulative weights (pre-swizzled B fragments)
    cumw_kernel<<<(DIN * DOUT + 255) / 256, 256, 0, stream>>>(W, bsw);

    // 3) per-support: GEMM then scatter
    const int gu = (NU + BM - 1) / BM;
    const int gv = (NI + BM - 1) / BM;
    const int ge = (E * 32 + 255) / 256;   // one wave per edge
    for (int s = 0; s < NS; ++s) {
        const float* Bs = bsw + (size_t)s * DIN * DOUT;
        gemm_wmma_f32<<<gu, 256, 0, stream>>>(U, Bs, tmp_u, NU);
        gemm_wmma_f32<<<gv, 256, 0, stream>>>(V, Bs, tmp_v, NI);
        spmm_kernel<<<ge, 256, 0, stream>>>(u_rows + (size_t)s * E, u_cols + (size_t)s * E,
                                            u_vals + (size_t)s * E, tmp_v, user_hidden, E);
        spmm_kernel<<<ge, 256, 0, stream>>>(i_rows + (size_t)s * E, i_cols + (size_t)s * E,
                                            i_vals + (size_t)s * E, tmp_u, item_hidden, E);
    }

    // 4) ReLU in place
    {
        size_t n4 = ((size_t)(NU + NI) * DOUT) / 4;
        int blocks = (int)((n4 + 255) / 256);
        if (blocks > 4096) blocks = 4096;
        relu_kernel<<<blocks, 256, 0, stream>>>(out, n4);
    }
}